// DomainCorrelation_17394617548903
// MI455X (gfx1250) — compile-verified
//
#include <hip/hip_runtime.h>

typedef __attribute__((ext_vector_type(2))) float v2f;
typedef __attribute__((ext_vector_type(8))) float v8f;

#define D_IN   256
#define D_PROJ 128
#define BATCH  4
#define NSEQ   4096

// -------------------------------------------------------------------------
// Kernel 1: fused Q/K projection GEMMs.
//   [B*N=16384, 256] @ [256, 128] + bias, 3 domains x {K,Q} = 6 GEMMs.
//   One wave per 16x16 output tile. The K=256 reduction runs as two
//   interleaved 128-deep WMMA chains (c0, c1) so consecutive
//   v_wmma_f32_16x16x4_f32 are independent (no accumulator RAW stall).
// -------------------------------------------------------------------------
__global__ __launch_bounds__(32)
void proj_kernel(const float* __restrict__ xs0,
                 const float* __restrict__ xs1,
                 const float* __restrict__ xs2,
                 const float* __restrict__ Wk, const float* __restrict__ bk,
                 const float* __restrict__ Wq, const float* __restrict__ bq,
                 float* __restrict__ Kbuf, float* __restrict__ Qbuf)
{
    const int tileM = blockIdx.x;            // 0..1023  (M = B*N = 16384)
    const int tileP = blockIdx.y;            // 0..7     (128 cols)
    const int mt    = blockIdx.z;            // 0..5     domain*2 + isQ
    const int domain = mt >> 1;
    const int isQ    = mt & 1;

    const float* x    = (domain == 0) ? xs0 : (domain == 1) ? xs1 : xs2;
    const float* W    = isQ ? Wq : Wk;
    const float* bias = isQ ? bq : bk;
    float* out = (isQ ? Qbuf : Kbuf) + (size_t)domain * (BATCH * NSEQ * D_PROJ);

    const int lane = threadIdx.x;
    const int half = lane >> 4;              // 0: K={0,1}, 1: K={2,3}
    const int l16  = lane & 15;
    const int m0   = tileM * 16;
    const int p0   = tileP * 16;
    const int kb   = half * 2;

    const float* arow = x + (size_t)(m0 + l16) * D_IN + kb;      // A row m0+l16
    const float* bcol = W + p0 + l16;                            // B col p0+l16

    v8f c0 = {}, c1 = {};
    #pragma unroll 4
    for (int s = 0; s < 32; ++s) {
        // chain 0: K-range [0,128)
        v2f a0 = *(const v2f*)(arow + 4 * s);
        v2f b0;
        b0.x = bcol[(size_t)(4 * s + kb)     * D_PROJ];
        b0.y = bcol[(size_t)(4 * s + kb + 1) * D_PROJ];
        // chain 1: K-range [128,256)
        v2f a1 = *(const v2f*)(arow + 4 * (s + 32));
        v2f b1;
        b1.x = bcol[(size_t)(4 * (s + 32) + kb)     * D_PROJ];
        b1.y = bcol[(size_t)(4 * (s + 32) + kb + 1) * D_PROJ];
        c0 = __builtin_amdgcn_wmma_f32_16x16x4_f32(false, a0, false, b0,
                                                   (short)0, c0, false, false);
        c1 = __builtin_amdgcn_wmma_f32_16x16x4_f32(false, a1, false, b1,
                                                   (short)0, c1, false, false);
    }

    const float bv = bias[p0 + l16];
    #pragma unroll
    for (int v = 0; v < 8; ++v) {
        // C/D layout: VGPR v -> row (v + 8*half), col l16
        out[(size_t)(m0 + v + 8 * half) * D_PROJ + p0 + l16] = (c0[v] + c1[v]) + bv;
    }
}

// -------------------------------------------------------------------------
// Kernel 2: attention-pair reduction (flash-softmax).
//   One wave per (32-query stripe, batch, pair): TWO 16x16 q-tiles share
//   every K-tile load -> 16 FLOP per L2 byte on the K stream and two
//   independent WMMA accumulator chains per B operand. Per-lane online
//   (rowmax, rowsumexp); mean(softmax_row) == rsum/rsum accumulated to acc.
// -------------------------------------------------------------------------
__global__ __launch_bounds__(32)
void attn_kernel(const float* __restrict__ Kbuf,
                 const float* __restrict__ Qbuf,
                 float* __restrict__ acc)
{
    static const int PI[6] = {0, 0, 0, 1, 1, 2};
    static const int PJ[6] = {0, 1, 2, 1, 2, 2};

    const int qt = blockIdx.x;               // 0..127  (stripe of 32 rows)
    const int b  = blockIdx.y;               // 0..3
    const int p  = blockIdx.z;               // 0..5
    const int di = PI[p], dj = PJ[p];

    const float* Q = Qbuf + ((size_t)dj * BATCH + b) * (NSEQ * D_PROJ);
    const float* K = Kbuf + ((size_t)di * BATCH + b) * (NSEQ * D_PROJ);

    const int lane = threadIdx.x;
    const int half = lane >> 4;
    const int l16  = lane & 15;
    const int kb   = half * 2;
    const int q0   = qt * 32;

    // Preload two Q tiles (A operands reused for every k-tile): 2 x 32 float2.
    v2f areg0[32], areg1[32];
    {
        const float* qrow0 = Q + (size_t)(q0 + l16) * D_PROJ + kb;
        const float* qrow1 = qrow0 + 16 * D_PROJ;
        #pragma unroll
        for (int s = 0; s < 32; ++s) {
            areg0[s] = *(const v2f*)(qrow0 + 4 * s);
            areg1[s] = *(const v2f*)(qrow1 + 4 * s);
        }
    }

    // Per-lane online softmax state: 8 rows per tile per lane.
    float rmax0[8], rsum0[8], rmax1[8], rsum1[8];
    #pragma unroll
    for (int v = 0; v < 8; ++v) {
        rmax0[v] = -3.0e38f; rsum0[v] = 0.0f;
        rmax1[v] = -3.0e38f; rsum1[v] = 0.0f;
    }

    for (int kt = 0; kt < NSEQ / 16; ++kt) {
        // B operand: transposed K tile; this lane's column = key kt*16+l16.
        const float* krow = K + (size_t)(kt * 16 + l16) * D_PROJ + kb;
        v8f c0 = {}, c1 = {};
        #pragma unroll 8
        for (int s = 0; s < 32; ++s) {
            v2f bvec = *(const v2f*)(krow + 4 * s);   // shared by both tiles
            c0 = __builtin_amdgcn_wmma_f32_16x16x4_f32(false, areg0[s], false, bvec,
                                                       (short)0, c0, false, false);
            c1 = __builtin_amdgcn_wmma_f32_16x16x4_f32(false, areg1[s], false, bvec,
                                                       (short)0, c1, false, false);
        }
        // Per-lane online (max, sumexp) update — no cross-lane traffic here.
        #pragma unroll
        for (int v = 0; v < 8; ++v) {
            float l0  = c0[v];
            float mn0 = fmaxf(rmax0[v], l0);
            rsum0[v]  = rsum0[v] * __expf(rmax0[v] - mn0) + __expf(l0 - mn0);
            rmax0[v]  = mn0;
            float l1  = c1[v];
            float mn1 = fmaxf(rmax1[v], l1);
            rsum1[v]  = rsum1[v] * __expf(rmax1[v] - mn1) + __expf(l1 - mn1);
            rmax1[v]  = mn1;
        }
    }

    // Cross-lane combine within each 16-lane half, then per-row
    // mean-of-softmax = rsum/rsum (exactly 1.0 in IEEE for finite s>0).
    float part = 0.0f;
    #pragma unroll
    for (int t = 0; t < 2; ++t) {
        #pragma unroll
        for (int v = 0; v < 8; ++v) {
            float m = t ? rmax1[v] : rmax0[v];
            float s = t ? rsum1[v] : rsum0[v];
            #pragma unroll
            for (int off = 1; off < 16; off <<= 1) {
                float mo = __shfl_xor(m, off, 32);
                float so = __shfl_xor(s, off, 32);
                float mn = fmaxf(m, mo);
                s = s * __expf(m - mn) + so * __expf(mo - mn);
                m = mn;
            }
            part += s / s;                   // sum_k softmax(row)[k]
        }
    }
    float other = __shfl_xor(part, 16, 32);  // the other half's rows
    if (lane == 0) atomicAdd(&acc[b * 6 + p], part + other);  // exact 32.0 adds
}

// -------------------------------------------------------------------------
// Kernel 3: finalize — corr matrix + three length-4 softmaxes (48 outputs).
// -------------------------------------------------------------------------
__global__ void zero_acc_kernel(float* __restrict__ acc)
{
    if (threadIdx.x < 24) acc[threadIdx.x] = 0.0f;
}

__global__ void finalize_kernel(const float* __restrict__ acc, float* __restrict__ out)
{
    if (threadIdx.x != 0 || blockIdx.x != 0) return;
    static const int PI[6] = {0, 0, 0, 1, 1, 2};
    static const int PJ[6] = {0, 1, 2, 1, 2, 2};
    const float inv = 1.0f / ((float)NSEQ * (float)NSEQ);

    float corr[BATCH][3][3];
    for (int b = 0; b < BATCH; ++b)
        for (int p = 0; p < 6; ++p) {
            float m = acc[b * 6 + p] * inv;
            corr[b][PI[p]][PJ[p]] = m;
            corr[b][PJ[p]][PI[p]] = m;
        }

    for (int b = 0; b < BATCH; ++b)
        for (int i = 0; i < 3; ++i)
            for (int j = 0; j < 3; ++j)
                out[b * 9 + i * 3 + j] = corr[b][i][j];

    for (int i = 0; i < 3; ++i) {
        float v[BATCH], mx = -3.0e38f;
        for (int b = 0; b < BATCH; ++b) {
            v[b] = (corr[b][i][0] + corr[b][i][1] + corr[b][i][2]) * (1.0f / 3.0f);
            mx = fmaxf(mx, v[b]);
        }
        float s = 0.0f;
        for (int b = 0; b < BATCH; ++b) { v[b] = __expf(v[b] - mx); s += v[b]; }
        for (int b = 0; b < BATCH; ++b) out[36 + i * 4 + b] = v[b] / s;
    }
}

// -------------------------------------------------------------------------
extern "C" void kernel_launch(void* const* d_in, const int* in_sizes, int n_in,
                              void* d_out, int out_size, void* d_ws, size_t ws_size,
                              hipStream_t stream)
{
    const float* spatial   = (const float*)d_in[0];
    const float* frequency = (const float*)d_in[1];
    const float* wavelet   = (const float*)d_in[2];
    const float* Wk        = (const float*)d_in[3];
    const float* bk        = (const float*)d_in[4];
    const float* Wq        = (const float*)d_in[5];
    const float* bq        = (const float*)d_in[6];
    float* out = (float*)d_out;

    // Workspace: K[3][B,N,128] | Q[3][B,N,128] | acc[24]  (~50.4 MB)
    float* Kbuf = (float*)d_ws;
    float* Qbuf = Kbuf + (size_t)3 * BATCH * NSEQ * D_PROJ;
    float* acc  = Qbuf + (size_t)3 * BATCH * NSEQ * D_PROJ;

    zero_acc_kernel<<<1, 32, 0, stream>>>(acc);
    proj_kernel<<<dim3(1024, 8, 6), 32, 0, stream>>>(spatial, frequency, wavelet,
                                                     Wk, bk, Wq, bq, Kbuf, Qbuf);
    attn_kernel<<<dim3(128, 4, 6), 32, 0, stream>>>(Kbuf, Qbuf, acc);
    finalize_kernel<<<1, 1, 0, stream>>>(acc, out);
}